// CombinedSARAFilter_30683246362700
// MI455X (gfx1250) — compile-verified
//
#include <hip/hip_runtime.h>
#include <cstdint>
#include <cstddef>

typedef __attribute__((ext_vector_type(2))) float v2f;
typedef __attribute__((ext_vector_type(8))) float v8f;

namespace {
constexpr int Bb = 8, Tt = 2000, Nc = 2048;
constexpr int SC_LEN = 80;              // timesteps per super-chunk
constexpr int NSC    = Tt / SC_LEN;     // 25 super-chunks
constexpr int NBLK   = SC_LEN / 16;     // 5 WMMA blocks per super-chunk
constexpr int NB16   = Nc / 16;         // 128 channel tiles
constexpr int STATE_STRIDE = Bb * NSC * Nc;

constexpr float INV_DT = 10.0f;
constexpr float K1f = 0.05f, K2f = 3.0f, K3f = 2.0f;
constexpr float FRA = 0.1f / 30.0f;     // dt/tau_RA
constexpr float FR  = 0.1f / 5.0f;      // dt/tau_r
constexpr float FD  = 0.1f / 30.0f;     // dt/tau_d

// recurrence in affine form:  r' = AR*r + BR*|dI| ; x' = AX*x + BX*u ; s' = AD*s + G*x'
constexpr double ARd = 1.0 - 0.1 / 30.0;
constexpr double AXd = 1.0 - 0.1 / 5.0;
constexpr double ADd = 1.0 - 0.1 / 30.0;
constexpr double Gd  = 0.1 / 30.0;
constexpr double BXd = 0.1 / 5.0;
constexpr double BRd = 2.0 * 0.1 / 30.0;

constexpr double dpow(double a, int n) { double r = 1.0; for (int i = 0; i < n; ++i) r *= a; return r; }
// P(j) = sum_{m=0}^{j} AD^(j-m) AX^m   (double-pole partial sum)
constexpr double PsD(int j) { double r = 0.0; for (int m = 0; m <= j; ++m) r += dpow(ADd, j - m) * dpow(AXd, m); return r; }

constexpr float ARf = (float)ARd, AXf = (float)AXd, ADf = (float)ADd;
constexpr float Gf = (float)Gd, BXf = (float)BXd, BRf = (float)BRd;
constexpr float AR16 = (float)dpow(ARd, 16), AX16 = (float)dpow(AXd, 16), AD16 = (float)dpow(ADd, 16);
constexpr float PHX16 = (float)(Gd * AXd * PsD(15));
constexpr float AR80 = (float)dpow(ARd, 80), AX80 = (float)dpow(AXd, 80), AD80 = (float)dpow(ADd, 80);
constexpr float PHX80 = (float)(Gd * AXd * PsD(79));
constexpr float INV_DAX = (float)(1.0 / (ADd - AXd));
constexpr float CSA = (float)(Gd * BXd / (ADd - AXd));
constexpr float AX2 = (float)(AXd * AXd), AD2 = (float)(ADd * ADd), AR2 = (float)(ARd * ARd);
} // namespace

__device__ __forceinline__ float fpowi(float a, int n) { float r = 1.f; for (int i = 0; i < n; ++i) r *= a; return r; }
// P(d) in float via closed form
__device__ __forceinline__ float psf(int d) { return (fpowi(ADf, d + 1) - fpowi(AXf, d + 1)) * INV_DAX; }

// ---------------- Pass 1: per-(chunk,channel) zero-init scan, record final state ----------------
__global__ __launch_bounds__(256) void sara_pass1(const float* __restrict__ in,
                                                  float* __restrict__ R, float* __restrict__ X,
                                                  float* __restrict__ S) {
  int tid = blockIdx.x * 256 + threadIdx.x;
  if (tid >= Bb * NSC * Nc) return;
  int n = tid % Nc; int r2 = tid / Nc; int sc = r2 % NSC; int b = r2 / NSC;
  size_t base = ((size_t)b * Tt + (size_t)sc * SC_LEN) * Nc + n;
  float prev = sc ? in[base - Nc] : 0.f;
  float r = 0.f, x = 0.f, s = 0.f;
  for (int t = 0; t < SC_LEN; ++t) {
    float cur = in[base + (size_t)t * Nc];
    float dI = (cur - prev) * INV_DT;
    r += (K3f * fabsf(dI) - r) * FRA;
    x += (K2f * dI + K1f * cur - x) * FR;
    s += (x - s) * FD;
    prev = cur;
  }
  int idx = (b * NSC + sc) * Nc + n;
  R[idx] = r; X[idx] = x; S[idx] = s;
}

// ---------------- Pass 2: combine chunk states -> incoming state per chunk (in place) ----------------
__global__ __launch_bounds__(256) void sara_pass2(float* __restrict__ R, float* __restrict__ X,
                                                  float* __restrict__ S) {
  int tid = blockIdx.x * 256 + threadIdx.x;
  if (tid >= Bb * Nc) return;
  int n = tid % Nc, b = tid / Nc;
  float rin = 0.f, xin = 0.f, sin_ = 0.f;
  for (int sc = 0; sc < NSC; ++sc) {
    int idx = (b * NSC + sc) * Nc + n;
    float rf = R[idx], xf = X[idx], sf = S[idx];
    R[idx] = rin; X[idx] = xin; S[idx] = sin_;   // incoming state for chunk sc
    float sn = AD80 * sin_ + PHX80 * xin + sf;   // homogeneous 80-step propagation + local final
    xin = AX80 * xin + xf;
    rin = AR80 * rin + rf;
    sin_ = sn;
  }
}

// ---------------- Pass 3: WMMA Toeplitz-conv scan, one wave = 16 channels x one super-chunk ----------------
__global__ __launch_bounds__(256) void sara_pass3(const float* __restrict__ in,
                                                  const float* __restrict__ wsR,
                                                  const float* __restrict__ wsX,
                                                  const float* __restrict__ wsS,
                                                  float* __restrict__ out) {
  __shared__ float lds[8][17 * 16];
  const int lane = threadIdx.x & 31;
  const int w = threadIdx.x >> 5;
  const int gw = blockIdx.x * 8 + w;
  const int nb = gw % NB16;
  int rest = gw / NB16;
  const int sc = rest % NSC;
  const int b = rest / NSC;
  const int n0 = nb * 16;
  const int M = lane & 15;      // output-time row (A/C) or channel (B/C)
  const int half = lane >> 4;

  // A-operand Toeplitz coefficients (16x4 tiles, K-chunk kk; ISA layout: v0={K0,K2}, v1={K1,K3})
  float aRAx[4], aRAy[4], aSAx[4], aSAy[4];
#pragma unroll
  for (int kk = 0; kk < 4; ++kk) {
    int K0 = 4 * kk + 2 * half;
    int d0 = M - K0, d1 = M - K0 - 1;
    aRAx[kk] = (d0 >= 0) ? BRf * fpowi(ARf, d0) : 0.f;
    aRAy[kk] = (d1 >= 0) ? BRf * fpowi(ARf, d1) : 0.f;
    aSAx[kk] = (d0 >= 0) ? Gf * BXf * psf(d0) : 0.f;
    aSAy[kk] = (d1 >= 0) ? Gf * BXf * psf(d1) : 0.f;
  }
  // Homogeneous-response matrix Phi (16x4): rows [phiR, phiX, phiS, 0]
  v2f aPhi;
  aPhi.x = half ? fpowi(ADf, M + 1) : fpowi(ARf, M + 1);  // v0: K0=phiR (half0), K2=phiS (half1)
  aPhi.y = half ? 0.f : Gf * AXf * psf(M);                // v1: K1=phiX (half0), K3=0 (half1)

  // incoming state, mirrored across the two lane-halves (lane c and c+16 read same address)
  const int sidx = (b * NSC + sc) * Nc + n0 + M;
  float Rst = wsR[sidx], Xst = wsX[sidx], Sst = wsS[sidx];

  float* tile = lds[w];
  for (int blk = 0; blk < NBLK; ++blk) {
    const int t0 = sc * SC_LEN + blk * 16;
    // stage rows t0-1 .. t0+15 (17 x 16 f32) into wave-private LDS; DS ops are in-order per wave
#pragma unroll
    for (int it = 0; it < 9; ++it) {
      int row = it * 2 + half;
      if (row < 17) {
        int t = t0 - 1 + row;
        float v = 0.f;
        if (t >= 0) v = in[((size_t)b * Tt + t) * Nc + n0 + M];
        tile[row * 16 + M] = v;
      }
    }
    // build B-operand drives: lane holds times t = 4kk + 2v + half for its channel M
    float bU[8], bA[8];
#pragma unroll
    for (int kk = 0; kk < 4; ++kk)
#pragma unroll
      for (int v = 0; v < 2; ++v) {
        int tl = 4 * kk + 2 * v + half;
        float It = tile[(tl + 1) * 16 + M];
        float Ip = tile[tl * 16 + M];
        float dI = (It - Ip) * INV_DT;
        bU[kk * 2 + v] = K2f * dI + K1f * It;
        bA[kk * 2 + v] = fabsf(dI);
      }
    // OUT(16t x 16ch) = Phi @ state + T_RA @ |dI| + T_SA @ u   (9 chained f32 WMMAs)
    v8f c = {0.f, 0.f, 0.f, 0.f, 0.f, 0.f, 0.f, 0.f};
    {
      v2f bSt; bSt.x = half ? Xst : Rst; bSt.y = half ? 0.f : Sst;  // B rows: [R;X;S;0]
      c = __builtin_amdgcn_wmma_f32_16x16x4_f32(false, aPhi, false, bSt, (short)0, c, false, false);
    }
#pragma unroll
    for (int kk = 0; kk < 4; ++kk) {
      v2f aA, bB;
      aA.x = aRAx[kk]; aA.y = aRAy[kk];
      bB.x = bA[kk * 2]; bB.y = bA[kk * 2 + 1];
      c = __builtin_amdgcn_wmma_f32_16x16x4_f32(false, aA, false, bB, (short)0, c, false, false);
      aA.x = aSAx[kk]; aA.y = aSAy[kk];
      bB.x = bU[kk * 2]; bB.y = bU[kk * 2 + 1];
      c = __builtin_amdgcn_wmma_f32_16x16x4_f32(false, aA, false, bB, (short)0, c, false, false);
    }
    // store C/D tile: VGPR v -> time row (v + 8*half), channel = lane%16
#pragma unroll
    for (int v = 0; v < 8; ++v)
      out[((size_t)b * Tt + (t0 + v + 8 * half)) * Nc + n0 + M] = c[v];

    // advance state by 16 steps: exponential-weighted folds over this lane's times, then xor-16 combine
    float accX = 0.f, accD = 0.f, accR = 0.f;
#pragma unroll
    for (int j = 0; j < 8; ++j) { accX = accX * AX2 + bU[j]; accD = accD * AD2 + bU[j]; accR = accR * AR2 + bA[j]; }
    if (!half) { accX *= AXf; accD *= ADf; accR *= ARf; }  // align weight a^(15-t)
    accX += __shfl_xor(accX, 16, 32);
    accD += __shfl_xor(accD, 16, 32);
    accR += __shfl_xor(accR, 16, 32);
    float Xn = AX16 * Xst + BXf * accX;
    float Rn = AR16 * Rst + BRf * accR;
    float Sn = AD16 * Sst + PHX16 * Xst + CSA * (ADf * accD - AXf * accX);
    Xst = Xn; Rst = Rn; Sst = Sn;
  }
}

extern "C" void kernel_launch(void* const* d_in, const int* in_sizes, int n_in,
                              void* d_out, int out_size, void* d_ws, size_t ws_size,
                              hipStream_t stream) {
  (void)in_sizes; (void)n_in; (void)out_size; (void)ws_size;
  const float* I = (const float*)d_in[0];
  float* out = (float*)d_out;
  float* ws = (float*)d_ws;
  float* wsR = ws;
  float* wsX = ws + STATE_STRIDE;
  float* wsS = ws + 2 * STATE_STRIDE;

  sara_pass1<<<(Bb * NSC * Nc) / 256, 256, 0, stream>>>(I, wsR, wsX, wsS);
  sara_pass2<<<(Bb * Nc) / 256, 256, 0, stream>>>(wsR, wsX, wsS);
  sara_pass3<<<(Bb * NSC * NB16) / 8, 256, 0, stream>>>(I, wsR, wsX, wsS, out);
}